// AlphaFoldLoss_9174050144605
// MI455X (gfx1250) — compile-verified
//
#include <hip/hip_runtime.h>
#include <hip/hip_bf16.h>

#define N_RES 768
#define DISTO_BINS 64
#define LDDT_BINS 50

typedef __attribute__((ext_vector_type(2))) float v2f;
typedef __attribute__((ext_vector_type(8))) float v8f;

__device__ __forceinline__ float wred_sum(float v) {
#pragma unroll
  for (int m = 16; m > 0; m >>= 1) v += __shfl_xor(v, m, 32);
  return v;
}
__device__ __forceinline__ float wred_max(float v) {
#pragma unroll
  for (int m = 16; m > 0; m >>= 1) v = fmaxf(v, __shfl_xor(v, m, 32));
  return v;
}

// ---------------- init: zero accumulators + score slots ----------------
__global__ void k_init(float* ws, int n) {
  int i = blockIdx.x * blockDim.x + threadIdx.x;
  if (i < n) ws[i] = 0.f;
}

// ---------------- mask denominators ----------------
// ws[1] = sum pseudo_beta_mask ; ws[3] = sum mask_ca ; ws[5] = sum atom_exists
__global__ void k_mask_sums(const float* __restrict__ pb_mask,
                            const float* __restrict__ aa_mask,
                            const float* __restrict__ exists,
                            float* ws) {
  float s1 = 0.f, s3 = 0.f, s5 = 0.f;
  for (int i = threadIdx.x; i < N_RES; i += blockDim.x) {
    s1 += pb_mask[i];
    s3 += aa_mask[i * 37 + 1];
  }
  for (int i = threadIdx.x; i < N_RES * 37; i += blockDim.x) s5 += exists[i];
  s1 = wred_sum(s1); s3 = wred_sum(s3); s5 = wred_sum(s5);
  __shared__ float sh[3][8];
  int lane = threadIdx.x & 31, w = threadIdx.x >> 5;
  if (lane == 0) { sh[0][w] = s1; sh[1][w] = s3; sh[2][w] = s5; }
  __syncthreads();
  if (threadIdx.x == 0) {
    float a = 0.f, b = 0.f, c = 0.f;
    for (int i = 0; i < 8; ++i) { a += sh[0][i]; b += sh[1][i]; c += sh[2][i]; }
    ws[1] = a; ws[3] = b; ws[5] = c;
  }
}

// ---------------- lddt score via WMMA f32 16x16x4 Gram tiles ----------------
// one wave (32 threads) per 16-row tile; loops over 48 column tiles.
// d2_ij = |x_i|^2 + |x_j|^2 - 2 * (X X^T)_ij
__global__ void __launch_bounds__(32)
k_lddt_score(const float* __restrict__ pred_pos,   // (768,37,3)
             const float* __restrict__ true_pos,   // (768,37,3)
             const float* __restrict__ aa_mask,    // (768,37)
             float* ws) {
  const int lane = threadIdx.x;
  const int ti   = blockIdx.x;       // row tile
  const int half = lane >> 4;        // 0 -> K{0,1}/M+0 ; 1 -> K{2,3}/M+8
  const int l16  = lane & 15;

  // A fragments (16x4, K=3 padded with 0): lane row = l16
  const int arow = ti * 16 + l16;
  const float* tA = true_pos + (size_t)arow * 37 * 3 + 3;  // CA atom (idx 1)
  const float* pA = pred_pos + (size_t)arow * 37 * 3 + 3;
  v2f aT, aP;
  aT.x = tA[half * 2]; aT.y = (half == 0) ? tA[1] : 0.f;
  aP.x = pA[half * 2]; aP.y = (half == 0) ? pA[1] : 0.f;

  // per-lane row data for the C-layout rows (VGPR v -> row ti*16+v+half*8)
  float rrT[8], rrP[8], mr[8];
#pragma unroll
  for (int v = 0; v < 8; ++v) {
    int r = ti * 16 + v + half * 8;
    const float* t = true_pos + (size_t)r * 37 * 3 + 3;
    const float* q = pred_pos + (size_t)r * 37 * 3 + 3;
    rrT[v] = t[0] * t[0] + t[1] * t[1] + t[2] * t[2];
    rrP[v] = q[0] * q[0] + q[1] * q[1] + q[2] * q[2];
    mr[v]  = aa_mask[r * 37 + 1];
  }

  float sScope[8], sWS[8];
#pragma unroll
  for (int v = 0; v < 8; ++v) { sScope[v] = 0.f; sWS[v] = 0.f; }

  for (int tj = 0; tj < N_RES / 16; ++tj) {
    const int bcol = tj * 16 + l16;
    const float* t = true_pos + (size_t)bcol * 37 * 3 + 3;
    const float* q = pred_pos + (size_t)bcol * 37 * 3 + 3;
    v2f bT, bP;
    bT.x = t[half * 2]; bT.y = (half == 0) ? t[1] : 0.f;
    bP.x = q[half * 2]; bP.y = (half == 0) ? q[1] : 0.f;
    const float rcT = t[0] * t[0] + t[1] * t[1] + t[2] * t[2];
    const float rcP = q[0] * q[0] + q[1] * q[1] + q[2] * q[2];
    const float mc  = aa_mask[bcol * 37 + 1];

    v8f cT = {}; v8f cP = {};
    cT = __builtin_amdgcn_wmma_f32_16x16x4_f32(false, aT, false, bT,
                                               (short)0, cT, false, false);
    cP = __builtin_amdgcn_wmma_f32_16x16x4_f32(false, aP, false, bP,
                                               (short)0, cP, false, false);
#pragma unroll
    for (int v = 0; v < 8; ++v) {
      const int row = ti * 16 + v + half * 8;
      float d2t = fmaxf(rrT[v] + rcT - 2.f * cT[v], 0.f);
      float d2p = fmaxf(rrP[v] + rcP - 2.f * cP[v], 0.f);
      float dt = sqrtf(1e-10f + d2t);
      float dp = sqrtf(1e-10f + d2p);
      float scope = ((dt < 15.f) ? 1.f : 0.f) * mr[v] * mc *
                    ((row != bcol) ? 1.f : 0.f);
      float l1 = fabsf(dt - dp);
      float sc = 0.25f * (((l1 < 0.5f) ? 1.f : 0.f) + ((l1 < 1.0f) ? 1.f : 0.f) +
                          ((l1 < 2.0f) ? 1.f : 0.f) + ((l1 < 4.0f) ? 1.f : 0.f));
      sScope[v] += scope;
      sWS[v]    += scope * sc;
    }
  }

  // reduce over columns: lanes 0..15 and 16..31 independently (xor 1,2,4,8)
#pragma unroll
  for (int v = 0; v < 8; ++v) {
    float a = sScope[v], b = sWS[v];
#pragma unroll
    for (int m = 8; m > 0; m >>= 1) {
      a += __shfl_xor(a, m, 32);
      b += __shfl_xor(b, m, 32);
    }
    if (l16 == 0) {
      int row = ti * 16 + v + half * 8;
      ws[6 + row] = (1e-10f + b) / (1e-10f + a);   // norm * (eps + sum ws)
    }
  }
}

// ---------------- lddt CE (50 bins, one wave per residue) ----------------
__global__ void k_lddt_ce(const float* __restrict__ logits,  // (768,50)
                          const float* __restrict__ aa_mask, // (768,37)
                          const float* __restrict__ score,   // ws+6
                          float* ws) {
  const int lane = threadIdx.x & 31;
  const int w = threadIdx.x >> 5;
  const int row = blockIdx.x * (blockDim.x >> 5) + w;
  float acc = 0.f;
  if (row < N_RES) {
    const float* lp = logits + (size_t)row * LDDT_BINS;
    const float NEG = -__builtin_inff();
    float v0 = (2 * lane + 0 < LDDT_BINS) ? lp[2 * lane + 0] : NEG;
    float v1 = (2 * lane + 1 < LDDT_BINS) ? lp[2 * lane + 1] : NEG;
    float mx = wred_max(fmaxf(v0, v1));
    float se = ((v0 != NEG) ? __expf(v0 - mx) : 0.f) +
               ((v1 != NEG) ? __expf(v1 - mx) : 0.f);
    se = wred_sum(se);
    float lse = mx + __logf(se);
    int bin = (int)floorf(score[row] * (float)LDDT_BINS);
    bin = min(max(bin, 0), LDDT_BINS - 1);
    float g0 = __shfl(v0, bin >> 1, 32);
    float g1 = __shfl(v1, bin >> 1, 32);
    float err = lse - ((bin & 1) ? g1 : g0);
    acc = err * aa_mask[row * 37 + 1];
  }
  __shared__ float sh[8];
  if (lane == 0) sh[w] = acc;
  __syncthreads();
  if (threadIdx.x == 0) {
    float s = 0.f;
    for (int i = 0; i < (int)(blockDim.x >> 5); ++i) s += sh[i];
    atomicAdd(&ws[2], s);
  }
}

// ---------------- distogram CE: streaming, one wave per pair ----------------
__global__ void k_disto(const float* __restrict__ logits,  // (768,768,64)
                        const float* __restrict__ beta,    // (768,3)
                        const float* __restrict__ mask,    // (768)
                        float* ws) {
  const int lane = threadIdx.x & 31;
  const int wv = blockIdx.x * (blockDim.x >> 5) + (threadIdx.x >> 5);
  const int nwv = gridDim.x * (blockDim.x >> 5);
  const int npairs = N_RES * N_RES;
  float acc = 0.f;

  for (int p = wv; p < npairs; p += nwv) {
    const int i = p / N_RES;
    const int j = p - i * N_RES;

    // prefetch next pair's logits (gfx1250 global_prefetch_b8)
    if (p + nwv < npairs)
      __builtin_prefetch(logits + (size_t)(p + nwv) * DISTO_BINS + 2 * lane, 0, 3);

    float dx = beta[i * 3 + 0] - beta[j * 3 + 0];
    float dy = beta[i * 3 + 1] - beta[j * 3 + 1];
    float dz = beta[i * 3 + 2] - beta[j * 3 + 2];
    float d2 = dx * dx + dy * dy + dz * dz;

    // true bin: count of 63 boundaries (2.3125 + 0.3125k)^2 below d2
    int cnt = 0;
    {
      float b0 = 2.3125f + 0.3125f * (float)(2 * lane);
      b0 *= b0;
      cnt += (d2 > b0) ? 1 : 0;
      if (2 * lane + 1 < 63) {
        float b1 = 2.3125f + 0.3125f * (float)(2 * lane + 1);
        b1 *= b1;
        cnt += (d2 > b1) ? 1 : 0;
      }
    }
#pragma unroll
    for (int m = 16; m > 0; m >>= 1) cnt += __shfl_xor(cnt, m, 32);
    const int bin = cnt;  // 0..63

    const float* lp = logits + (size_t)p * DISTO_BINS;
    v2f v = *(const v2f*)(lp + 2 * lane);
    float mx = wred_max(fmaxf(v.x, v.y));
    float se = __expf(v.x - mx) + __expf(v.y - mx);
    se = wred_sum(se);
    float lse = mx + __logf(se);

    float g0 = __shfl(v.x, bin >> 1, 32);
    float g1 = __shfl(v.y, bin >> 1, 32);
    float err = (lse - ((bin & 1) ? g1 : g0)) * mask[i] * mask[j];
    if (lane == 0) acc += err;
  }

  __shared__ float sh[8];
  const int w = threadIdx.x >> 5;
  if (lane == 0) sh[w] = acc;
  __syncthreads();
  if (threadIdx.x == 0) {
    float s = 0.f;
    for (int i = 0; i < (int)(blockDim.x >> 5); ++i) s += sh[i];
    atomicAdd(&ws[0], s);
  }
}

// ---------------- experimentally-resolved BCE ----------------
__global__ void k_exp(const float* __restrict__ logits,  // (768,37)
                      const float* __restrict__ aa_mask, // (768,37)
                      const float* __restrict__ exists,  // (768,37)
                      float* ws) {
  const int n = N_RES * 37;
  float acc = 0.f;
  for (int i = blockIdx.x * blockDim.x + threadIdx.x; i < n;
       i += gridDim.x * blockDim.x) {
    float x = logits[i], m = aa_mask[i], e = exists[i];
    float c = log1pf(__expf(-fabsf(x)));
    float sp_pos = fmaxf(x, 0.f) + c;    // -log_sigmoid(-x)
    float sp_neg = fmaxf(-x, 0.f) + c;   // -log_sigmoid(x)
    acc += e * (m * sp_neg + (1.f - m) * sp_pos);
  }
  acc = wred_sum(acc);
  __shared__ float sh[8];
  const int lane = threadIdx.x & 31, w = threadIdx.x >> 5;
  if (lane == 0) sh[w] = acc;
  __syncthreads();
  if (threadIdx.x == 0) {
    float s = 0.f;
    for (int i = 0; i < (int)(blockDim.x >> 5); ++i) s += sh[i];
    atomicAdd(&ws[4], s);
  }
}

// ---------------- final combine ----------------
__global__ void k_final(const float* __restrict__ ws,
                        const float* __restrict__ resolution,
                        float* out) {
  if (threadIdx.x == 0 && blockIdx.x == 0) {
    float res = resolution[0];
    float gate = (res >= 0.1f && res <= 3.0f) ? 1.f : 0.f;
    float l_disto = ws[0] / (1e-6f + ws[1] * ws[1]);
    float l_plddt = gate * ws[2] / (1e-10f + ws[3]);
    float l_exp   = gate * ws[4] / (1e-8f + ws[5]);
    out[0] = 0.3f * l_disto + 0.01f * l_plddt + 0.01f * l_exp;
  }
}

extern "C" void kernel_launch(void* const* d_in, const int* in_sizes, int n_in,
                              void* d_out, int out_size, void* d_ws, size_t ws_size,
                              hipStream_t stream) {
  const float* disto_logits = (const float*)d_in[0];
  const float* pseudo_beta  = (const float*)d_in[1];
  const float* pb_mask      = (const float*)d_in[2];
  const float* lddt_logits  = (const float*)d_in[3];
  const float* pred_pos     = (const float*)d_in[4];
  const float* true_pos     = (const float*)d_in[5];
  const float* aa_mask      = (const float*)d_in[6];
  const float* exp_logits   = (const float*)d_in[7];
  const float* exists       = (const float*)d_in[8];
  const float* resolution   = (const float*)d_in[9];
  float* out = (float*)d_out;
  float* ws  = (float*)d_ws;

  const int ws_floats = 6 + N_RES;
  k_init<<<(ws_floats + 255) / 256, 256, 0, stream>>>(ws, ws_floats);
  k_mask_sums<<<1, 256, 0, stream>>>(pb_mask, aa_mask, exists, ws);
  k_lddt_score<<<N_RES / 16, 32, 0, stream>>>(pred_pos, true_pos, aa_mask, ws);
  k_lddt_ce<<<N_RES / 8, 256, 0, stream>>>(lddt_logits, aa_mask, ws + 6, ws);
  k_disto<<<1024, 256, 0, stream>>>(disto_logits, pseudo_beta, pb_mask, ws);
  k_exp<<<112, 256, 0, stream>>>(exp_logits, aa_mask, exists, ws);
  k_final<<<1, 32, 0, stream>>>(ws, resolution, out);
}